// SpGQAFlashDecodeAttention_31997506355745
// MI455X (gfx1250) — compile-verified
//
#include <hip/hip_runtime.h>
#include <hip/hip_bf16.h>

typedef __attribute__((ext_vector_type(2))) float v2f;
typedef __attribute__((ext_vector_type(8))) float v8f;

#define R_    8
#define B_    8
#define H_    32
#define KVH_  8
#define G_    4
#define D_    128
#define DV_   128
#define P_    2048
#define NW    8          // waves per workgroup
#define TILE  16
#define NTILES (P_ / TILE)

#define SCALE_F 0.08838834764831845f   // 1/sqrt(128)
#define NEG_BIG (-1e30f)

__device__ __forceinline__ float shflf(float v, int srcLane) {
  return __int_as_float(__builtin_amdgcn_ds_bpermute(srcLane << 2, __float_as_int(v)));
}
__device__ __forceinline__ int shfli(int v, int srcLane) {
  return __builtin_amdgcn_ds_bpermute(srcLane << 2, v);
}
__device__ __forceinline__ v8f wmma4(v2f a, v2f b, v8f c) {
  // D = A(16x4 f32) * B(4x16 f32) + C(16x16 f32); exact fp32 path on CDNA5
  return __builtin_amdgcn_wmma_f32_16x16x4_f32(
      /*neg_a=*/false, a, /*neg_b=*/false, b,
      /*c_mod=*/(short)0, c, /*reuse_a=*/false, /*reuse_b=*/false);
}

// ---------------------------------------------------------------------------
// Kernel 1: per-(rank,batch,kvhead) flash-decode partial attention.
// grid = R*B*KVH blocks, 256 threads (8 wave32 waves).
// ---------------------------------------------------------------------------
__global__ __launch_bounds__(256) void fdecode_partial(
    const float* __restrict__ q,           // [B,H,D]
    const float* __restrict__ k_cache,     // [R,P,1,KVH,D]
    const float* __restrict__ v_cache,     // [R,P,1,KVH,DV]
    const int*   __restrict__ kv_lens,     // [R,B]
    const int*   __restrict__ block_table, // [R,B,P]
    float*       __restrict__ pout,        // [R,B,KVH,G,DV]
    float*       __restrict__ lse_ws)      // [R,B,KVH,G]
{
  __shared__ float q_lds[G_ * D_];
  __shared__ float p_lds[NW][TILE * 16];
  __shared__ float m_lds[NW][G_];
  __shared__ float l_lds[NW][G_];
  __shared__ float acc_lds[NW][G_][DV_];

  const int bid  = blockIdx.x;
  const int kvh  = bid % KVH_;
  const int b    = (bid / KVH_) % B_;
  const int r    = bid / (KVH_ * B_);
  const int tid  = threadIdx.x;
  const int lane = tid & 31;
  const int wid  = tid >> 5;
  const int half = lane >> 4;     // 0: lanes 0-15, 1: lanes 16-31
  const int l15  = lane & 15;

  // ---- stage Q (G x D) for this (b,kvh) into LDS ----
  const float* qg = q + ((size_t)b * H_ + (size_t)kvh * G_) * D_;
  for (int i = tid; i < G_ * D_; i += 256) q_lds[i] = qg[i];
  __syncthreads();

  // ---- build Q^T B-fragments (4-channel chunks), zero pad heads >= G ----
  v2f qf[D_ / 4];
  {
    const int h = l15;
#pragma unroll
    for (int cc = 0; cc < D_ / 4; ++cc) {
      const int chan = cc * 4 + half * 2;
      float x = 0.f, y = 0.f;
      if (h < G_) { x = q_lds[h * D_ + chan]; y = q_lds[h * D_ + chan + 1]; }
      qf[cc].x = x;
      qf[cc].y = y;
    }
  }

  const int    kvlen = kv_lens[r * B_ + b];
  const int*   bt    = block_table + ((size_t)r * B_ + b) * P_;
  const float* kbase = k_cache + (size_t)r * P_ * KVH_ * D_;
  const float* vbase = v_cache + (size_t)r * P_ * KVH_ * DV_;

  float m_run = NEG_BIG, l_run = 0.f;
  v8f acc[DV_ / 16];
#pragma unroll
  for (int c = 0; c < DV_ / 16; ++c)
#pragma unroll
    for (int k = 0; k < 8; ++k) acc[c][k] = 0.f;

  // ---- main token-tile loop (uniform trip count per wave; kvlen >= 128
  //      guarantees each wave's first tile is fully valid) ----
  for (int tile = wid; tile < NTILES; tile += NW) {
    const int base = tile * TILE;
    if (base >= kvlen) break;

    const int tok  = base + l15;
    const int page = bt[tok];                       // paged gather (page_size=1)
    const float* krow = kbase + ((size_t)page * KVH_ + kvh) * D_ + half * 2;

    // ---- S = K_tile * Q^T : [token x head], 32 WMMAs across D=128 ----
    v8f s;
#pragma unroll
    for (int k = 0; k < 8; ++k) s[k] = 0.f;
#pragma unroll
    for (int cc = 0; cc < D_ / 4; ++cc) {
      v2f a = *(const v2f*)(krow + cc * 4);         // b64 per-lane gather
      s = wmma4(a, qf[cc], s);
    }

    // ---- scale + length mask (token = VGPR idx + 8*half) ----
#pragma unroll
    for (int i = 0; i < 8; ++i) {
      const int t = base + i + half * 8;
      s[i] = (t < kvlen) ? s[i] * SCALE_F : NEG_BIG;
    }

    // ---- online softmax; per-head state lives at head = lane%16 ----
    float tm = s[0];
#pragma unroll
    for (int i = 1; i < 8; ++i) tm = fmaxf(tm, s[i]);
    tm = fmaxf(tm, shflf(tm, lane ^ 16));
    const float m_new = fmaxf(m_run, tm);
    const float alpha = expf(m_run - m_new);
    float psum = 0.f;
#pragma unroll
    for (int i = 0; i < 8; ++i) {
      const float p = expf(s[i] - m_new);
      s[i] = p;
      psum += p;
    }
    psum += shflf(psum, lane ^ 16);
    l_run = l_run * alpha + psum;
    m_run = m_new;

    // ---- rescale accumulator: C VGPR j holds head j (lo) / j+8 (hi) ----
    float ab[8];
#pragma unroll
    for (int j = 0; j < 8; ++j) ab[j] = shflf(alpha, j + half * 8);
#pragma unroll
    for (int c = 0; c < DV_ / 16; ++c)
#pragma unroll
      for (int j = 0; j < 8; ++j) acc[c][j] *= ab[j];

    // ---- stage P into per-wave LDS tile [token][head] for transpose ----
#pragma unroll
    for (int i = 0; i < 8; ++i)
      p_lds[wid][(i + half * 8) * 16 + l15] = s[i];
    __builtin_amdgcn_wave_barrier();                // keep ds store->load order

    // ---- O += P^T * V : A = P^T (head x 4 tokens), B = V (4 tokens x 16 dv)
#pragma unroll
    for (int k0 = 0; k0 < TILE; k0 += 4) {
      const int rk = k0 + half * 2;
      v2f af;
      af.x = p_lds[wid][rk * 16 + l15];
      af.y = p_lds[wid][(rk + 1) * 16 + l15];
      const int pg0 = shfli(page, rk);              // page id of tile row rk
      const int pg1 = shfli(page, rk + 1);
      const float* vr0 = vbase + ((size_t)pg0 * KVH_ + kvh) * DV_;
      const float* vr1 = vbase + ((size_t)pg1 * KVH_ + kvh) * DV_;
#pragma unroll
      for (int c = 0; c < DV_ / 16; ++c) {
        const int dv = c * 16 + l15;
        v2f bf;
        bf.x = vr0[dv];                             // half-wave 64B contiguous
        bf.y = vr1[dv];
        acc[c] = wmma4(af, bf, acc[c]);
      }
    }
  }

  // ---- publish per-wave flash state, merge across waves ----
  if (lane < G_) { m_lds[wid][lane] = m_run; l_lds[wid][lane] = l_run; }
  if (lane < 16) {
#pragma unroll
    for (int c = 0; c < DV_ / 16; ++c)
#pragma unroll
      for (int j = 0; j < G_; ++j)
        acc_lds[wid][j][c * 16 + lane] = acc[c][j];
  }
  __syncthreads();

  float* po = pout   + (size_t)bid * G_ * DV_;
  float* lo = lse_ws + (size_t)bid * G_;
  for (int o = tid; o < G_ * DV_; o += 256) {
    const int g  = o >> 7;
    const int dv = o & (DV_ - 1);
    float mt = NEG_BIG;
#pragma unroll
    for (int w = 0; w < NW; ++w) mt = fmaxf(mt, m_lds[w][g]);
    float L = 0.f, val = 0.f;
#pragma unroll
    for (int w = 0; w < NW; ++w) {
      const float e = expf(m_lds[w][g] - mt);
      L   += e * l_lds[w][g];
      val += e * acc_lds[w][g][dv];
    }
    po[o] = val / L;
    if (dv == 0) lo[g] = mt + logf(L);
  }
}

// ---------------------------------------------------------------------------
// Kernel 2: cross-rank online-softmax combine (reference's lse merge).
// ---------------------------------------------------------------------------
__global__ __launch_bounds__(256) void fdecode_combine(
    const float* __restrict__ pout,    // [R,B,KVH,G,DV]
    const float* __restrict__ lse_ws,  // [R,B,KVH,G]
    float*       __restrict__ out)     // [B,H,DV]
{
  const int idx = blockIdx.x * 256 + threadIdx.x;
  if (idx >= B_ * KVH_ * G_ * DV_) return;
  const int dv  = idx & (DV_ - 1);
  const int g   = (idx >> 7) & (G_ - 1);
  const int kvh = (idx >> 9) & (KVH_ - 1);
  const int b   = idx >> 12;

  float ls[R_];
  float mx = NEG_BIG;
#pragma unroll
  for (int r = 0; r < R_; ++r) {
    ls[r] = lse_ws[(((size_t)r * B_ + b) * KVH_ + kvh) * G_ + g];
    mx = fmaxf(mx, ls[r]);
  }
  float S = 0.f, val = 0.f;
#pragma unroll
  for (int r = 0; r < R_; ++r) {
    const float e = expf(ls[r] - mx);
    S   += e;
    val += e * pout[((((size_t)r * B_ + b) * KVH_ + kvh) * G_ + g) * DV_ + dv];
  }
  out[((size_t)b * H_ + (size_t)kvh * G_ + g) * DV_ + dv] = val / S;
}

// ---------------------------------------------------------------------------
extern "C" void kernel_launch(void* const* d_in, const int* in_sizes, int n_in,
                              void* d_out, int out_size, void* d_ws, size_t ws_size,
                              hipStream_t stream) {
  const float* q    = (const float*)d_in[0];
  const float* kc   = (const float*)d_in[1];
  const float* vc   = (const float*)d_in[2];
  const int*   lens = (const int*)d_in[3];
  const int*   bt   = (const int*)d_in[4];

  float* pout = (float*)d_ws;                                   // 1 MiB partials
  float* lse  = pout + (size_t)R_ * B_ * KVH_ * G_ * DV_;       // + 8 KiB lse

  fdecode_partial<<<R_ * B_ * KVH_, 256, 0, stream>>>(q, kc, vc, lens, bt, pout, lse);

  const int n2 = B_ * KVH_ * G_ * DV_;
  fdecode_combine<<<(n2 + 255) / 256, 256, 0, stream>>>(pout, lse, (float*)d_out);
}